// SMAL_80152679678534
// MI455X (gfx1250) — compile-verified
//
#include <hip/hip_runtime.h>
#include <math.h>

#define NBATCH 1024
#define NVERT  3889
#define NJ     33
#define NBETA  10
#define NP     288            // (NJ-1)*9
#define N3     (NVERT * 3)    // 11667
#define NLD    11680          // N3 padded to multiple of 16

typedef float v2f __attribute__((ext_vector_type(2)));
typedef float v8f __attribute__((ext_vector_type(8)));

// ---------------------------------------------------------------------------
// Kernel 1: Rodrigues rotation matrices + pose features (rot - I for j>=1)
// ---------------------------------------------------------------------------
__global__ void k_rodrigues(const float* __restrict__ go,   // [B,3]
                            const float* __restrict__ bp,   // [B,(J-1)*3]
                            float* __restrict__ rot,        // [B,J,9]
                            float* __restrict__ pf) {       // [B,288]
    int idx = blockIdx.x * blockDim.x + threadIdx.x;
    if (idx >= NBATCH * NJ) return;
    int j = idx % NJ, b = idx / NJ;
    float rx, ry, rz;
    if (j == 0) {
        rx = go[b * 3 + 0]; ry = go[b * 3 + 1]; rz = go[b * 3 + 2];
    } else {
        const float* p = bp + (size_t)b * (NJ - 1) * 3 + (j - 1) * 3;
        rx = p[0]; ry = p[1]; rz = p[2];
    }
    // reference: angle = ||r + 1e-8||, axis = r / angle
    float ax = rx + 1e-8f, ay = ry + 1e-8f, az = rz + 1e-8f;
    float ang = sqrtf(ax * ax + ay * ay + az * az);
    float inv = 1.0f / ang;
    float x = rx * inv, y = ry * inv, z = rz * inv;
    float s = sinf(ang), c = cosf(ang), t = 1.0f - c;
    float R[9];
    R[0] = 1.0f - t * (y * y + z * z);
    R[1] = -s * z + t * x * y;
    R[2] =  s * y + t * x * z;
    R[3] =  s * z + t * x * y;
    R[4] = 1.0f - t * (x * x + z * z);
    R[5] = -s * x + t * y * z;
    R[6] = -s * y + t * x * z;
    R[7] =  s * x + t * y * z;
    R[8] = 1.0f - t * (x * x + y * y);
    float* ro = rot + (size_t)idx * 9;
#pragma unroll
    for (int i = 0; i < 9; ++i) ro[i] = R[i];
    if (j >= 1) {
        float* po = pf + (size_t)b * NP + (j - 1) * 9;
        po[0] = R[0] - 1.0f; po[1] = R[1];        po[2] = R[2];
        po[3] = R[3];        po[4] = R[4] - 1.0f; po[5] = R[5];
        po[6] = R[6];        po[7] = R[7];        po[8] = R[8] - 1.0f;
    }
}

// ---------------------------------------------------------------------------
// Kernel 2: batch-independent joint regression precompute
//   jpre[j][c][l] (l<10) = sum_v Jreg[j,v] * shapedirs[v,c,l]
//   jpre[j][c][10]       = sum_v Jreg[j,v] * v_template[v,c]
// ---------------------------------------------------------------------------
__global__ __launch_bounds__(256) void k_jpre(const float* __restrict__ JR,  // [J,V]
                                              const float* __restrict__ vt,  // [V,3]
                                              const float* __restrict__ sd,  // [V,3,10]
                                              float* __restrict__ jpre) {    // [J,33]
    const int j = blockIdx.x;
    float acc[33];
#pragma unroll
    for (int i = 0; i < 33; ++i) acc[i] = 0.0f;
    for (int v = threadIdx.x; v < NVERT; v += 256) {
        float r = JR[(size_t)j * NVERT + v];
#pragma unroll
        for (int c = 0; c < 3; ++c) {
            acc[c * 11 + 10] += r * vt[v * 3 + c];
            const float* sdr = sd + (size_t)v * 30 + c * 10;
#pragma unroll
            for (int l = 0; l < 10; ++l) acc[c * 11 + l] += r * sdr[l];
        }
    }
#pragma unroll
    for (int i = 0; i < 33; ++i)
        for (int off = 16; off > 0; off >>= 1)
            acc[i] += __shfl_down(acc[i], off, 32);
    __shared__ float wsum[8][33];
    int wid = threadIdx.x >> 5, lane = threadIdx.x & 31;
    if (lane == 0)
        for (int i = 0; i < 33; ++i) wsum[wid][i] = acc[i];
    __syncthreads();
    if (threadIdx.x < 33) {
        float s = 0.0f;
        for (int w = 0; w < 8; ++w) s += wsum[w][threadIdx.x];
        jpre[j * 33 + threadIdx.x] = s;
    }
}

// ---------------------------------------------------------------------------
// Kernel 3: joints[b,j,c] = jpre[j][c][10] + sum_l jpre[j][c][l] * betas[b,l]
// ---------------------------------------------------------------------------
__global__ void k_joints(const float* __restrict__ jpre,
                         const float* __restrict__ betas,   // [B,10]
                         float* __restrict__ joints) {      // [B,J,3]
    int idx = blockIdx.x * blockDim.x + threadIdx.x;
    if (idx >= NBATCH * NJ * 3) return;
    int c = idx % 3;
    int j = (idx / 3) % NJ;
    int b = idx / (3 * NJ);
    const float* row = jpre + j * 33 + c * 11;
    const float* be = betas + b * NBETA;
    float s = row[10];
#pragma unroll
    for (int l = 0; l < 10; ++l) s += row[l] * be[l];
    joints[idx] = s;
}

// ---------------------------------------------------------------------------
// Kernel 4: forward kinematics (one thread per body; general parents tree)
//   chain[b,j] = 3x4 world transform; A[b,j] = chain with rest-pose removed
// ---------------------------------------------------------------------------
__global__ void k_fk(const float* __restrict__ rot,      // [B,J,9]
                     const float* __restrict__ joints,   // [B,J,3]
                     const int* __restrict__ parents,    // [J]
                     float* __restrict__ chain,          // [B,J,12]
                     float* __restrict__ Amat) {         // [B,J,12]
    int b = blockIdx.x * blockDim.x + threadIdx.x;
    if (b >= NBATCH) return;
    const float* Rb = rot + (size_t)b * NJ * 9;
    const float* Jb = joints + (size_t)b * NJ * 3;
    float* Cb = chain + (size_t)b * NJ * 12;
    float* Ab = Amat + (size_t)b * NJ * 12;
    // root: G = [R0 | j0]
    {
        float G[12];
#pragma unroll
        for (int r = 0; r < 3; ++r) {
            G[r * 4 + 0] = Rb[r * 3 + 0];
            G[r * 4 + 1] = Rb[r * 3 + 1];
            G[r * 4 + 2] = Rb[r * 3 + 2];
            G[r * 4 + 3] = Jb[r];
        }
#pragma unroll
        for (int q = 0; q < 12; ++q) Cb[q] = G[q];
#pragma unroll
        for (int r = 0; r < 3; ++r) {
            float rj = G[r * 4 + 0] * Jb[0] + G[r * 4 + 1] * Jb[1] + G[r * 4 + 2] * Jb[2];
            Ab[r * 4 + 0] = G[r * 4 + 0];
            Ab[r * 4 + 1] = G[r * 4 + 1];
            Ab[r * 4 + 2] = G[r * 4 + 2];
            Ab[r * 4 + 3] = G[r * 4 + 3] - rj;
        }
    }
    for (int i = 1; i < NJ; ++i) {
        int p = parents[i];
        float Gp[12];
#pragma unroll
        for (int q = 0; q < 12; ++q) Gp[q] = Cb[p * 12 + q];   // same-thread RAW, in-order
        const float* Ri = Rb + i * 9;
        float rel[3];
#pragma unroll
        for (int c = 0; c < 3; ++c) rel[c] = Jb[i * 3 + c] - Jb[p * 3 + c];
        float G[12];
#pragma unroll
        for (int r = 0; r < 3; ++r) {
#pragma unroll
            for (int c = 0; c < 3; ++c)
                G[r * 4 + c] = Gp[r * 4 + 0] * Ri[0 * 3 + c] +
                               Gp[r * 4 + 1] * Ri[1 * 3 + c] +
                               Gp[r * 4 + 2] * Ri[2 * 3 + c];
            G[r * 4 + 3] = Gp[r * 4 + 0] * rel[0] + Gp[r * 4 + 1] * rel[1] +
                           Gp[r * 4 + 2] * rel[2] + Gp[r * 4 + 3];
        }
#pragma unroll
        for (int q = 0; q < 12; ++q) Cb[i * 12 + q] = G[q];
#pragma unroll
        for (int r = 0; r < 3; ++r) {
            float rj = G[r * 4 + 0] * Jb[i * 3 + 0] + G[r * 4 + 1] * Jb[i * 3 + 1] +
                       G[r * 4 + 2] * Jb[i * 3 + 2];
            Ab[i * 12 + r * 4 + 0] = G[r * 4 + 0];
            Ab[i * 12 + r * 4 + 1] = G[r * 4 + 1];
            Ab[i * 12 + r * 4 + 2] = G[r * 4 + 2];
            Ab[i * 12 + r * 4 + 3] = G[r * 4 + 3] - rj;
        }
    }
}

// ---------------------------------------------------------------------------
// Kernel 5: pose-blendshape GEMM via V_WMMA_F32_16X16X4_F32
//   PO[m, n] = sum_k PF[m, k] * PD[k, n],  M=1024, K=288, N=11667 (pad 11680)
// One wave -> one 16x16 tile, 72 WMMA steps over K.
// A 16x4 f32 layout:  lanes 0-15 hold {K=0,K=1}, lanes 16-31 hold {K=2,K=3}
// B 4x16 f32 layout:  VGPR0 = rows K=0(l0-15)/K=2(l16-31), VGPR1 = K=1/K=3
// D 16x16 f32 layout: VGPR r -> M=r (lanes 0-15), M=r+8 (lanes 16-31)
// ---------------------------------------------------------------------------
__global__ __launch_bounds__(256) void k_gemm_pose(const float* __restrict__ PF,  // [1024,288]
                                                   const float* __restrict__ PD,  // [288,11667]
                                                   float* __restrict__ PO) {      // [1024,11680]
    const int wid = blockIdx.x * 8 + (threadIdx.x >> 5);
    const int lane = threadIdx.x & 31;
    const int mTile = wid & 63;          // 64 tiles over M=1024
    const int nTile = wid >> 6;          // 730 tiles over N
    const int m0 = mTile << 4;
    const int n0 = nTile << 4;
    const int half = lane >> 4;
    const int l16 = lane & 15;
    const int m = m0 + l16;
    const int n = n0 + l16;
    const int nc = (n < N3) ? n : (N3 - 1);   // clamp loads; padded cols masked by stride
    const float* pfrow = PF + (size_t)m * NP;
    v8f acc = {};
    for (int k = 0; k < NP; k += 4) {
        const int kk = k + 2 * half;
        v2f a, bvec;
        a.x = pfrow[kk];
        a.y = pfrow[kk + 1];
        bvec.x = PD[(size_t)kk * N3 + nc];
        bvec.y = PD[(size_t)(kk + 1) * N3 + nc];
        acc = __builtin_amdgcn_wmma_f32_16x16x4_f32(false, a, false, bvec,
                                                    (short)0, acc, false, false);
    }
    float* out = PO + (size_t)(m0 + 8 * half) * NLD + n;
#pragma unroll
    for (int r = 0; r < 8; ++r)
        out[(size_t)r * NLD] = acc[r];
}

// ---------------------------------------------------------------------------
// Kernel 6: fused skinning.  v_posed recomputed (v_template + shapedirs*betas
// + pose offset), per-vertex blend of 33 joint transforms (A staged in LDS),
// homogeneous transform + transl.
// ---------------------------------------------------------------------------
__global__ __launch_bounds__(256) void k_skin(const float* __restrict__ betas,   // [B,10]
                                              const float* __restrict__ transl,  // [B,3]
                                              const float* __restrict__ vt,      // [V,3]
                                              const float* __restrict__ sd,      // [V,3,10]
                                              const float* __restrict__ LW,      // [V,J]
                                              const float* __restrict__ Amat,    // [B,J,12]
                                              const float* __restrict__ PO,      // [B,11680]
                                              float* __restrict__ out) {         // [B,V,3]
    __shared__ float As[NJ * 12];
    __shared__ float bs[NBETA];
    __shared__ float ts[3];
    const int b = blockIdx.y;
    for (int i = threadIdx.x; i < NJ * 12; i += blockDim.x)
        As[i] = Amat[(size_t)b * NJ * 12 + i];
    if (threadIdx.x < NBETA) bs[threadIdx.x] = betas[b * NBETA + threadIdx.x];
    if (threadIdx.x < 3) ts[threadIdx.x] = transl[b * 3 + threadIdx.x];
    __syncthreads();
    const int v = blockIdx.x * blockDim.x + threadIdx.x;
    if (v >= NVERT) return;
    float vp[3];
#pragma unroll
    for (int c = 0; c < 3; ++c) {
        float x = vt[v * 3 + c];
        const float* sdr = sd + (size_t)v * 30 + c * 10;
#pragma unroll
        for (int l = 0; l < NBETA; ++l) x += sdr[l] * bs[l];
        x += PO[(size_t)b * NLD + v * 3 + c];
        vp[c] = x;
    }
    float T[12];
#pragma unroll
    for (int i = 0; i < 12; ++i) T[i] = 0.0f;
    const float* wr = LW + (size_t)v * NJ;
    for (int j = 0; j < NJ; ++j) {
        float w = wr[j];
#pragma unroll
        for (int i = 0; i < 12; ++i) T[i] += w * As[j * 12 + i];
    }
    float* o = out + ((size_t)b * NVERT + v) * 3;
#pragma unroll
    for (int c = 0; c < 3; ++c)
        o[c] = T[c * 4 + 0] * vp[0] + T[c * 4 + 1] * vp[1] +
               T[c * 4 + 2] * vp[2] + T[c * 4 + 3] + ts[c];
}

// ---------------------------------------------------------------------------
// Launch
// ---------------------------------------------------------------------------
extern "C" void kernel_launch(void* const* d_in, const int* in_sizes, int n_in,
                              void* d_out, int out_size, void* d_ws, size_t ws_size,
                              hipStream_t stream) {
    const float* betas     = (const float*)d_in[0];
    const float* go        = (const float*)d_in[1];
    const float* bp        = (const float*)d_in[2];
    const float* transl    = (const float*)d_in[3];
    const float* vtemplate = (const float*)d_in[4];
    const float* shapedirs = (const float*)d_in[5];
    const float* posedirs  = (const float*)d_in[6];
    const float* Jreg      = (const float*)d_in[7];
    const float* lbsw      = (const float*)d_in[8];
    const int*   parents   = (const int*)d_in[9];
    float* verts = (float*)d_out;

    // workspace layout (bytes, 256-aligned)
    constexpr size_t ALN = 256;
    auto align_up = [](size_t x) { return (x + ALN - 1) & ~(ALN - 1); };
    size_t off = 0;
    size_t off_rot   = off; off += align_up((size_t)NBATCH * NJ * 9 * 4);
    size_t off_pf    = off; off += align_up((size_t)NBATCH * NP * 4);
    size_t off_jpre  = off; off += align_up((size_t)NJ * 33 * 4);
    size_t off_jnt   = off; off += align_up((size_t)NBATCH * NJ * 3 * 4);
    size_t off_chain = off; off += align_up((size_t)NBATCH * NJ * 12 * 4);
    size_t off_A     = off; off += align_up((size_t)NBATCH * NJ * 12 * 4);
    size_t off_po    = off; off += align_up((size_t)NBATCH * NLD * 4);
    (void)ws_size; (void)n_in; (void)in_sizes; (void)out_size;

    char* ws = (char*)d_ws;
    float* rot   = (float*)(ws + off_rot);
    float* pf    = (float*)(ws + off_pf);
    float* jpre  = (float*)(ws + off_jpre);
    float* jnt   = (float*)(ws + off_jnt);
    float* chain = (float*)(ws + off_chain);
    float* Amat  = (float*)(ws + off_A);
    float* po    = (float*)(ws + off_po);

    // 1. rotations + pose features
    {
        int n = NBATCH * NJ;
        k_rodrigues<<<(n + 255) / 256, 256, 0, stream>>>(go, bp, rot, pf);
    }
    // 2. batch-independent joint regressor folding
    k_jpre<<<NJ, 256, 0, stream>>>(Jreg, vtemplate, shapedirs, jpre);
    // 3. joints
    {
        int n = NBATCH * NJ * 3;
        k_joints<<<(n + 255) / 256, 256, 0, stream>>>(jpre, betas, jnt);
    }
    // 4. forward kinematics
    k_fk<<<(NBATCH + 255) / 256, 256, 0, stream>>>(rot, jnt, parents, chain, Amat);
    // 5. pose-blendshape GEMM (WMMA): 64*730 = 46720 waves, 8 waves/block
    k_gemm_pose<<<46720 / 8, 256, 0, stream>>>(pf, posedirs, po);
    // 6. fused skinning
    {
        dim3 grid((NVERT + 255) / 256, NBATCH);
        k_skin<<<grid, 256, 0, stream>>>(betas, transl, vtemplate, shapedirs,
                                         lbsw, Amat, po, verts);
    }
}